// SparseGCN_90984587198551
// MI455X (gfx1250) — compile-verified
//
#include <hip/hip_runtime.h>
#include <stdint.h>

#define U_NODES 100000
#define I_NODES 50000
#define NODES   (U_NODES + I_NODES)
#define DIM     64
#define NNZ     4800000
#define LAYERS  3

#define SPMM_BLOCKS 2048
#define SPMM_TPB    256
#define WAVES_PER_BLOCK (SPMM_TPB / 32)

// Low 32 bits of a generic pointer to a __shared__ object == LDS byte offset
// (ISA: LDS aperture address truncates to addr[31:0], wave-LDS-relative).
__device__ __forceinline__ uint32_t lds_byte_off(const void* p) {
    return (uint32_t)(uintptr_t)p;
}

// Stage one 32-edge tile into LDS via async DMA (ASYNCcnt path).
// AoS layout, 16 B per edge: {row, col, val, pad} so the consumer can read
// one edge with a single 16-byte-aligned ds_load_b128 broadcast.
__device__ __forceinline__ void stage_async_load(uint32_t lds_base, int e0, int lane,
                                                 const int* __restrict__ erow,
                                                 const int* __restrict__ ecol,
                                                 const float* __restrict__ eval) {
    uint64_t ar = (uint64_t)(uintptr_t)(erow + e0 + lane);
    uint64_t ac = (uint64_t)(uintptr_t)(ecol + e0 + lane);
    uint64_t av = (uint64_t)(uintptr_t)(eval + e0 + lane);
    uint32_t l0 = lds_base + (uint32_t)(lane * 16);       // row
    uint32_t l1 = l0 + 4u;                                // col
    uint32_t l2 = l0 + 8u;                                // val
    asm volatile("global_load_async_to_lds_b32 %0, %1, off" :: "v"(l0), "v"(ar) : "memory");
    asm volatile("global_load_async_to_lds_b32 %0, %1, off" :: "v"(l1), "v"(ac) : "memory");
    asm volatile("global_load_async_to_lds_b32 %0, %1, off" :: "v"(l2), "v"(av) : "memory");
}

// emb0 = concat(user, item); write it to out_lo (first output), seed the
// accumulator (second output region) and the layer-0 propagation buffer.
__global__ void init_kernel(const float4* __restrict__ user,
                            const float4* __restrict__ item,
                            float4* __restrict__ out_lo,
                            float4* __restrict__ acc,
                            float4* __restrict__ x0) {
    int i = blockIdx.x * blockDim.x + threadIdx.x;     // over NODES*DIM/4
    const int total = NODES * (DIM / 4);
    if (i >= total) return;
    const int ulim = U_NODES * (DIM / 4);
    float4 e = (i < ulim) ? user[i] : item[i - ulim];
    out_lo[i] = e;
    acc[i]    = e;
    x0[i]     = e;
}

// COO scatter SpMM: one wave per edge, lane -> 2 dims. Edge triples are
// double-buffered through LDS via async DMA; per edge the wave does one
// ds_load_b128 broadcast + one coalesced 8B gather + 2 f32 atomics per lane.
// x (38.4 MB) and y (38.4 MB) are L2-resident (192 MB L2), so gathers and
// atomic adds never touch HBM; the only HBM stream is the edge arrays.
__global__ __launch_bounds__(SPMM_TPB)
void spmm_scatter_kernel(const int* __restrict__ erow,
                         const int* __restrict__ ecol,
                         const float* __restrict__ eval,
                         const float* __restrict__ x,
                         float* __restrict__ y) {
    // per wave: 2 buffers x 32 edges x 16 bytes
    __shared__ uint32_t stage[WAVES_PER_BLOCK][2][128];

    const int lane = threadIdx.x & 31;
    const int wib  = threadIdx.x >> 5;
    const int wid  = blockIdx.x * WAVES_PER_BLOCK + wib;
    const int nw   = gridDim.x * WAVES_PER_BLOCK;
    const int S    = NNZ / 32;                         // 150000 stages, exact

    const uint32_t ldsA = lds_byte_off(&stage[wib][0][0]);
    const uint32_t ldsB = lds_byte_off(&stage[wib][1][0]);

    int s   = wid;
    int cur = 0;
    if (s < S) stage_async_load(ldsA, s * 32, lane, erow, ecol, eval);

    while (s < S) {
        const int snext = s + nw;
        if (snext < S) {
            stage_async_load(cur ? ldsA : ldsB, snext * 32, lane, erow, ecol, eval);
            // 3 newest (next stage) may stay in flight; async loads complete
            // in order, so cnt<=3 means this stage's 3 DMAs have landed.
            asm volatile("s_wait_asynccnt 0x3" ::: "memory");
        } else {
            asm volatile("s_wait_asynccnt 0x0" ::: "memory");
        }

        const uint32_t* buf = &stage[wib][cur][0];

        #pragma unroll 4
        for (int e = 0; e < 32; ++e) {
            // one 16B-aligned LDS broadcast read per edge (row, col, val, pad)
            const int4 t = *(const int4*)(buf + e * 4);
            const int   r = t.x;
            const int   c = t.y;
            const float v = __int_as_float(t.z);
            const float2 xv = *(const float2*)(x + (size_t)c * DIM + (lane << 1));
            float* yr = y + (size_t)r * DIM + (lane << 1);
            __hip_atomic_fetch_add(yr,     v * xv.x, __ATOMIC_RELAXED, __HIP_MEMORY_SCOPE_AGENT);
            __hip_atomic_fetch_add(yr + 1, v * xv.y, __ATOMIC_RELAXED, __HIP_MEMORY_SCOPE_AGENT);
        }

        s = snext;
        cur ^= 1;
    }
}

// acc = (acc + y) * s   (s = 1 for layers 0..1, 0.25 fused on the last layer)
__global__ void add_scale_kernel(float4* __restrict__ acc,
                                 const float4* __restrict__ yv,
                                 float s, int total) {
    int i = blockIdx.x * blockDim.x + threadIdx.x;
    if (i >= total) return;
    float4 a = acc[i];
    float4 b = yv[i];
    a.x = (a.x + b.x) * s;
    a.y = (a.y + b.y) * s;
    a.z = (a.z + b.z) * s;
    a.w = (a.w + b.w) * s;
    acc[i] = a;
}

extern "C" void kernel_launch(void* const* d_in, const int* in_sizes, int n_in,
                              void* d_out, int out_size, void* d_ws, size_t ws_size,
                              hipStream_t stream) {
    (void)in_sizes; (void)n_in; (void)out_size; (void)ws_size;

    const float* user = (const float*)d_in[0];
    const float* item = (const float*)d_in[1];
    const float* eval = (const float*)d_in[2];
    const int*   erow = (const int*)d_in[3];
    const int*   ecol = (const int*)d_in[4];

    float* out_lo = (float*)d_out;                       // emb0
    float* acc    = out_lo + (size_t)NODES * DIM;        // light_out accumulator
    float* buf0   = (float*)d_ws;                        // ping
    float* buf1   = buf0 + (size_t)NODES * DIM;          // pong

    const int total4 = NODES * (DIM / 4);
    const int eltBlocks = (total4 + 255) / 256;

    init_kernel<<<eltBlocks, 256, 0, stream>>>((const float4*)user, (const float4*)item,
                                               (float4*)out_lo, (float4*)acc, (float4*)buf0);

    float* xb = buf0;
    float* yb = buf1;
    for (int l = 0; l < LAYERS; ++l) {
        hipMemsetAsync(yb, 0, (size_t)NODES * DIM * sizeof(float), stream);
        spmm_scatter_kernel<<<SPMM_BLOCKS, SPMM_TPB, 0, stream>>>(erow, ecol, eval, xb, yb);
        const float sc = (l == LAYERS - 1) ? 0.25f : 1.0f;
        add_scale_kernel<<<eltBlocks, 256, 0, stream>>>((float4*)acc, (const float4*)yb, sc, total4);
        float* t = xb; xb = yb; yb = t;
    }
}